// SimpleGNN_40931038331301
// MI455X (gfx1250) — compile-verified
//
#include <hip/hip_runtime.h>

#define F 64           // F_IN == HIDDEN == 64
#define NCLS 16
#define NGRAPH 64
#define WS_STRIDE 68   // padded transposed-W stride: (68*col+k)%64 = (4col+k)%64 -> conflict-free b64 fragment loads

typedef float v2f __attribute__((ext_vector_type(2)));
typedef float v8f __attribute__((ext_vector_type(8)));

// ---------------- degree ----------------
__global__ void k_deg_init(float* __restrict__ deg, int n) {
  int i = blockIdx.x * blockDim.x + threadIdx.x;
  if (i < n) deg[i] = 1.0f;   // self-loop
}

__global__ void k_deg_count(const int* __restrict__ dst, float* __restrict__ deg, int e) {
  int i = blockIdx.x * blockDim.x + threadIdx.x;
  if (i < e) atomicAdd(&deg[dst[i]], 1.0f);
}

__global__ void k_deg_rsqrt(float* __restrict__ deg, int n) {
  int i = blockIdx.x * blockDim.x + threadIdx.x;
  if (i < n) deg[i] = rsqrtf(deg[i]);
}

// ---------------- node GEMM: T[n x 64] = X[n x 64] @ W[64 x 64] ----------------
// 256 threads = 8 waves; each wave computes a 16-row x 64-col tile with
// v_wmma_f32_16x16x4_f32, K=64 in 16 steps, 4 column-tile accumulators.
// W is staged TRANSPOSED in LDS so each B fragment {W[k][n],W[k+1][n]} is one
// contiguous, bank-conflict-free ds_load_b64.
__global__ __launch_bounds__(256) void k_gemm64(const float* __restrict__ X,
                                                const float* __restrict__ W,
                                                float* __restrict__ T, int nrows) {
  __shared__ float Ws[F * WS_STRIDE];
  // cooperative transpose-stage: Ws[col][k] = W[k][col]
  for (int i = threadIdx.x; i < F * F; i += 256) {
    int k = i >> 6, col = i & 63;
    Ws[col * WS_STRIDE + k] = W[i];
  }
  __syncthreads();

  const int wave = threadIdx.x >> 5;
  const int lane = threadIdx.x & 31;
  const int tile = blockIdx.x * 8 + wave;
  if (tile * 16 >= nrows) return;          // wave-uniform: EXEC all-ones for WMMA

  const int row0 = tile * 16;
  const int mn   = lane & 15;              // M for A, N for B/D
  const int koff = (lane >> 4) << 1;       // 0 or 2

  const float* __restrict__ xrow = X + (size_t)(row0 + mn) * F + koff;
  const float* __restrict__ wc0  = Ws + (size_t)mn * WS_STRIDE + koff;

  v8f acc0 = {}, acc1 = {}, acc2 = {}, acc3 = {};
#pragma unroll 4
  for (int k0 = 0; k0 < F; k0 += 4) {
    v2f a  = *(const v2f*)(xrow + k0);
    v2f b0 = *(const v2f*)(wc0 + k0 + 0  * WS_STRIDE);
    v2f b1 = *(const v2f*)(wc0 + k0 + 16 * WS_STRIDE);
    v2f b2 = *(const v2f*)(wc0 + k0 + 32 * WS_STRIDE);
    v2f b3 = *(const v2f*)(wc0 + k0 + 48 * WS_STRIDE);
    acc0 = __builtin_amdgcn_wmma_f32_16x16x4_f32(false, a, false, b0, (short)0, acc0, false, false);
    acc1 = __builtin_amdgcn_wmma_f32_16x16x4_f32(false, a, false, b1, (short)0, acc1, false, false);
    acc2 = __builtin_amdgcn_wmma_f32_16x16x4_f32(false, a, false, b2, (short)0, acc2, false, false);
    acc3 = __builtin_amdgcn_wmma_f32_16x16x4_f32(false, a, false, b3, (short)0, acc3, false, false);
  }

  // D layout: VGPR v -> row v + 8*(lane>=16), col = lane&15 (+16 per col tile)
  const int rbase = row0 + ((lane >> 4) << 3);
#pragma unroll
  for (int v = 0; v < 8; ++v) {
    float* orow = T + (size_t)(rbase + v) * F + mn;
    orow[0]  = acc0[v];
    orow[16] = acc1[v];
    orow[32] = acc2[v];
    orow[48] = acc3[v];
  }
}

// ---------------- agg = T * invdeg + bias (self-loop + bias init) ----------------
__global__ void k_selfinit(const float* __restrict__ T, const float* __restrict__ isq,
                           const float* __restrict__ bias, float* __restrict__ agg, int n) {
  int i = blockIdx.x * blockDim.x + threadIdx.x;   // over n*64
  if (i < n * F) {
    int node = i >> 6;
    int f    = i & 63;
    float s  = isq[node];
    agg[i] = T[i] * (s * s) + bias[f];
  }
}

// ---------------- edge scatter: agg[dst] += T[src] * isq[src]*isq[dst] ----------------
// 16 threads per edge, float4 chunks; gathers hit L2 (feature matrix is 25.6 MB).
__global__ void k_scatter(const float* __restrict__ T, const int* __restrict__ src,
                          const int* __restrict__ dst, const float* __restrict__ isq,
                          float* __restrict__ agg, int e) {
  int idx  = blockIdx.x * blockDim.x + threadIdx.x;
  int edge = idx >> 4;
  if (edge >= e) return;
  int c = idx & 15;
  int s = src[edge];
  int d = dst[edge];
  float coef = isq[s] * isq[d];
  float4 v = *(const float4*)(T + (size_t)s * F + c * 4);
  float* out = agg + (size_t)d * F + c * 4;
  atomicAdd(out + 0, v.x * coef);
  atomicAdd(out + 1, v.y * coef);
  atomicAdd(out + 2, v.z * coef);
  atomicAdd(out + 3, v.w * coef);
}

__global__ void k_relu(float* __restrict__ a, int n64) {
  int i = blockIdx.x * blockDim.x + threadIdx.x;
  if (i < n64) a[i] = fmaxf(a[i], 0.0f);
}

// ---------------- pooling ----------------
__global__ void k_pool_zero(float* __restrict__ pooled, float* __restrict__ counts) {
  int i = blockIdx.x * blockDim.x + threadIdx.x;
  if (i < NGRAPH * F) pooled[i] = 0.0f;
  if (i < NGRAPH) counts[i] = 0.0f;
}

__global__ void k_pool(const float* __restrict__ h, const int* __restrict__ batch,
                       float* __restrict__ pooled, float* __restrict__ counts, int n) {
  int idx  = blockIdx.x * blockDim.x + threadIdx.x;
  int node = idx >> 4;
  if (node >= n) return;
  int c = idx & 15;
  int g = batch[node];
  float4 v = *(const float4*)(h + (size_t)node * F + c * 4);
  float* out = pooled + g * F + c * 4;
  atomicAdd(out + 0, v.x);
  atomicAdd(out + 1, v.y);
  atomicAdd(out + 2, v.z);
  atomicAdd(out + 3, v.w);
  if (c == 0) atomicAdd(&counts[g], 1.0f);
}

__global__ void k_pool_norm(float* __restrict__ pooled, const float* __restrict__ counts) {
  int i = blockIdx.x * blockDim.x + threadIdx.x;
  if (i < NGRAPH * F) pooled[i] /= fmaxf(counts[i >> 6], 1.0f);
}

// ---------------- final GEMM: out[64 x 16] = P[64 x 64] @ Wlin[64 x 16] + blin ----------------
// one block, 4 waves, each wave a 16x16 tile via WMMA.
__global__ __launch_bounds__(128) void k_final(const float* __restrict__ P,
                                               const float* __restrict__ Wl,
                                               const float* __restrict__ bl,
                                               float* __restrict__ out) {
  const int wave = threadIdx.x >> 5;
  const int lane = threadIdx.x & 31;
  const int mn   = lane & 15;
  const int koff = (lane >> 4) << 1;
  const int row0 = wave * 16;

  v8f acc = {};
#pragma unroll 4
  for (int k0 = 0; k0 < F; k0 += 4) {
    v2f a = *(const v2f*)(P + (row0 + mn) * F + k0 + koff);
    v2f b = {Wl[(k0 + koff) * NCLS + mn], Wl[(k0 + koff + 1) * NCLS + mn]};
    acc = __builtin_amdgcn_wmma_f32_16x16x4_f32(false, a, false, b, (short)0, acc, false, false);
  }
  const int rbase = row0 + ((lane >> 4) << 3);
#pragma unroll
  for (int v = 0; v < 8; ++v)
    out[(rbase + v) * NCLS + mn] = acc[v] + bl[mn];
}

extern "C" void kernel_launch(void* const* d_in, const int* in_sizes, int n_in,
                              void* d_out, int out_size, void* d_ws, size_t ws_size,
                              hipStream_t stream) {
  const float* x     = (const float*)d_in[0];
  const int*   eidx  = (const int*)d_in[1];    // [2, E] int32
  const int*   batch = (const int*)d_in[2];
  // d_in[3] = num_graphs (device scalar; grid is static so use NGRAPH=64)
  const float* W1    = (const float*)d_in[4];
  const float* b1    = (const float*)d_in[5];
  const float* W2    = (const float*)d_in[6];
  const float* b2    = (const float*)d_in[7];
  const float* Wlin  = (const float*)d_in[8];
  const float* blin  = (const float*)d_in[9];
  float* out = (float*)d_out;

  const int n = in_sizes[0] / F;
  const int e = in_sizes[1] / 2;
  const int* src = eidx;
  const int* dst = eidx + e;

  // workspace carve-out
  float* ws     = (float*)d_ws;
  float* bufA   = ws;                          // n*64  (GEMM result T)
  float* bufB   = bufA + (size_t)n * F;        // n*64  (agg / layer output)
  float* deg    = bufB + (size_t)n * F;        // n     (deg -> rsqrt(deg))
  float* pooled = deg + n;                     // 64*64
  float* counts = pooled + NGRAPH * F;         // 64

  const int T256   = 256;
  const int gN     = (n + T256 - 1) / T256;
  const int gE     = (e + T256 - 1) / T256;
  const int gNF    = (n * F + T256 - 1) / T256;
  const int gE16   = (int)(((long long)e * 16 + T256 - 1) / T256);
  const int gN16   = (int)(((long long)n * 16 + T256 - 1) / T256);
  const int ntiles = (n + 15) / 16;
  const int gGemm  = (ntiles + 7) / 8;

  // symmetric-normalization coefficients (shared by both layers)
  k_deg_init <<<gN, T256, 0, stream>>>(deg, n);
  k_deg_count<<<gE, T256, 0, stream>>>(dst, deg, e);
  k_deg_rsqrt<<<gN, T256, 0, stream>>>(deg, n);

  // ---- layer 1 ----
  k_gemm64  <<<gGemm, 256, 0, stream>>>(x, W1, bufA, n);
  k_selfinit<<<gNF, T256, 0, stream>>>(bufA, deg, b1, bufB, n);
  k_scatter <<<gE16, T256, 0, stream>>>(bufA, src, dst, deg, bufB, e);
  k_relu    <<<gNF, T256, 0, stream>>>(bufB, n * F);

  // ---- layer 2 (ping-pong buffers) ----
  k_gemm64  <<<gGemm, 256, 0, stream>>>(bufB, W2, bufA, n);
  k_selfinit<<<gNF, T256, 0, stream>>>(bufA, deg, b2, bufB, n);
  k_scatter <<<gE16, T256, 0, stream>>>(bufA, src, dst, deg, bufB, e);
  k_relu    <<<gNF, T256, 0, stream>>>(bufB, n * F);

  // ---- global mean pool + classifier ----
  k_pool_zero<<<(NGRAPH * F + T256 - 1) / T256, T256, 0, stream>>>(pooled, counts);
  k_pool     <<<gN16, T256, 0, stream>>>(bufB, batch, pooled, counts, n);
  k_pool_norm<<<(NGRAPH * F + T256 - 1) / T256, T256, 0, stream>>>(pooled, counts);
  k_final    <<<1, 128, 0, stream>>>(pooled, Wlin, blin, out);
}